// GraphAttentionLayer_65936337928657
// MI455X (gfx1250) — compile-verified
//
#include <hip/hip_runtime.h>
#include <hip/hip_bf16.h>

typedef float v2f __attribute__((ext_vector_type(2)));
typedef float v8f __attribute__((ext_vector_type(8)));

#define GAT_B     8
#define GAT_N     2048
#define GAT_FIN   128
#define GAT_FOUT  64
#define GAT_ALPHA 0.2f
#define NEG_BIG   (-9.0e15f)

// ---------------------------------------------------------------------------
// Kernel 1: Wh[b,n,:] = h[b,n,:] @ W   (M=16384, K=128, N=64)
// One wave per 16-row tile. f32 WMMA 16x16x4.
// A frag (16x4): lane L -> row M=L%16, VGPR v -> K = k0 + v + 2*(L/16)
// B frag (4x16): lane L -> col N=L%16, VGPR v -> K = k0 + v + 2*(L/16)
// C/D (16x16):   lane L -> col N=L%16, VGPR v -> row M = v + 8*(L/16)
// ---------------------------------------------------------------------------
__global__ __launch_bounds__(128) void gat_wh_kernel(
    const float* __restrict__ h, const float* __restrict__ W,
    float* __restrict__ Wh) {
  const int wave = blockIdx.x * (blockDim.x >> 5) + (threadIdx.x >> 5);
  const int lane = threadIdx.x & 31;
  const int half = lane >> 4;
  const int l16  = lane & 15;
  const int m0   = wave * 16;            // row tile in flattened (B*N)

  v8f c[4] = {v8f{}, v8f{}, v8f{}, v8f{}};
  const float* hrow = h + (size_t)(m0 + l16) * GAT_FIN;

  for (int k0 = 0; k0 < GAT_FIN; k0 += 4) {
    const int ka = k0 + 2 * half;
    v2f a;
    a.x = hrow[ka];
    a.y = hrow[ka + 1];
    const float* w0 = W + (size_t)ka * GAT_FOUT;
#pragma unroll
    for (int t = 0; t < 4; ++t) {
      v2f b;
      b.x = w0[t * 16 + l16];
      b.y = w0[GAT_FOUT + t * 16 + l16];
      c[t] = __builtin_amdgcn_wmma_f32_16x16x4_f32(
          false, a, false, b, (short)0, c[t], false, false);
    }
  }

#pragma unroll
  for (int t = 0; t < 4; ++t)
#pragma unroll
    for (int v = 0; v < 8; ++v)
      Wh[(size_t)(m0 + v + 8 * half) * GAT_FOUT + t * 16 + l16] = c[t][v];
}

// ---------------------------------------------------------------------------
// Kernel 2: e1[row] = Wh[row,:] . a[0:64], e2[row] = Wh[row,:] . a[64:128]
// ---------------------------------------------------------------------------
__global__ __launch_bounds__(256) void gat_e_kernel(
    const float* __restrict__ Wh, const float* __restrict__ avec,
    float* __restrict__ e1, float* __restrict__ e2) {
  const int row = blockIdx.x * blockDim.x + threadIdx.x;  // 0..B*N-1
  const float* w = Wh + (size_t)row * GAT_FOUT;
  float s1 = 0.f, s2 = 0.f;
#pragma unroll 8
  for (int o = 0; o < GAT_FOUT; ++o) {
    const float v = w[o];
    s1 += v * avec[o];
    s2 += v * avec[GAT_FOUT + o];
  }
  e1[row] = s1;
  e2[row] = s2;
}

// ---------------------------------------------------------------------------
// Kernel 3: fused flash-style masked softmax + P@Wh + ELU.
// One wave per 16-row i-tile. j-loop in chunks of 16.
// Row-stat layout: per-lane value for row = lane%16 (replicated in both
// 16-lane halves after shfl_xor(16)). Accumulator rows are v + 8*half, so
// per-v scales are fetched with a per-lane shuffle (ds_bpermute).
// A-fragment of P: lane L holds P[L%16][4*ks + v + 2*half] for v=0,1.
// ---------------------------------------------------------------------------
__global__ __launch_bounds__(128) void gat_attn_kernel(
    const int* __restrict__ adj, const float* __restrict__ Wh,
    const float* __restrict__ e1g, const float* __restrict__ e2g,
    float* __restrict__ out) {
  const int wave = blockIdx.x * (blockDim.x >> 5) + (threadIdx.x >> 5);
  const int lane = threadIdx.x & 31;
  const int half = lane >> 4;
  const int l16  = lane & 15;

  const int tilesPerBatch = GAT_N / 16;        // 128
  const int b  = wave / tilesPerBatch;
  const int i0 = (wave % tilesPerBatch) * 16;

  const int*   adjb = adj + (size_t)b * GAT_N * GAT_N;
  const float* Whb  = Wh  + (size_t)b * GAT_N * GAT_FOUT;
  const float* e2b  = e2g + (size_t)b * GAT_N;

  const float ei = e1g[(size_t)b * GAT_N + i0 + l16];   // row-stat (row=l16)
  const int* arow = adjb + (size_t)(i0 + l16) * GAT_N;

  v8f  c[4] = {v8f{}, v8f{}, v8f{}, v8f{}};
  float m = -3.0e38f;   // running row max   (row-stat)
  float l = 0.0f;       // running row sum   (row-stat)

  for (int j0 = 0; j0 < GAT_N; j0 += 16) {
    __builtin_prefetch(arow + j0 + 16, 0, 1);   // global_prefetch_b8 next chunk

    // --- build masked leaky-relu'd logits: 8 per lane, row = l16 ----------
    float p[8];
    float emax = -3.0e38f;
#pragma unroll
    for (int ks = 0; ks < 4; ++ks) {
      const int c0 = 4 * ks + 2 * half;                 // even -> int2 aligned
      const int2 av = *(const int2*)(arow + j0 + c0);
      float ex = ei + e2b[j0 + c0];
      float ey = ei + e2b[j0 + c0 + 1];
      ex = ex > 0.f ? ex : GAT_ALPHA * ex;
      ey = ey > 0.f ? ey : GAT_ALPHA * ey;
      ex = av.x > 0 ? ex : NEG_BIG;
      ey = av.y > 0 ? ey : NEG_BIG;
      p[2 * ks]     = ex;
      p[2 * ks + 1] = ey;
      emax = fmaxf(emax, fmaxf(ex, ey));
    }
    emax = fmaxf(emax, __shfl_xor(emax, 16, 32));       // full 16-col row max

    // --- online softmax update -------------------------------------------
    const float mnew  = fmaxf(m, emax);
    const float scale = __expf(m - mnew);               // row-stat rescale
    m = mnew;

    float rsum = 0.f;
#pragma unroll
    for (int q = 0; q < 8; ++q) {
      p[q] = __expf(p[q] - mnew);
      rsum += p[q];
    }
    rsum += __shfl_xor(rsum, 16, 32);
    l = l * scale + rsum;

    // per-accumulator-row scales: acc row r = v + 8*half, stat lives in lane r
    float sc[8];
#pragma unroll
    for (int v = 0; v < 8; ++v) sc[v] = __shfl(scale, v + 8 * half, 32);
#pragma unroll
    for (int t = 0; t < 4; ++t)
#pragma unroll
      for (int v = 0; v < 8; ++v) c[t][v] *= sc[v];

    // --- accumulate P @ Wh with f32 WMMA: 4 K-steps x 4 N-tiles ----------
#pragma unroll
    for (int ks = 0; ks < 4; ++ks) {
      v2f a;
      a.x = p[2 * ks];
      a.y = p[2 * ks + 1];
      const float* w0 = Whb + (size_t)(j0 + 4 * ks + 2 * half) * GAT_FOUT;
#pragma unroll
      for (int t = 0; t < 4; ++t) {
        v2f bb;
        bb.x = w0[t * 16 + l16];
        bb.y = w0[GAT_FOUT + t * 16 + l16];
        c[t] = __builtin_amdgcn_wmma_f32_16x16x4_f32(
            false, a, false, bb, (short)0, c[t], false, false);
      }
    }
  }

  // --- normalize by row sum, ELU, store ----------------------------------
  const float inv = 1.0f / l;                           // row-stat
  float invv[8];
#pragma unroll
  for (int v = 0; v < 8; ++v) invv[v] = __shfl(inv, v + 8 * half, 32);

#pragma unroll
  for (int t = 0; t < 4; ++t)
#pragma unroll
    for (int v = 0; v < 8; ++v) {
      float x = c[t][v] * invv[v];
      x = x > 0.f ? x : (__expf(x) - 1.0f);             // ELU (alpha=1)
      out[((size_t)b * GAT_N + i0 + v + 8 * half) * GAT_FOUT + t * 16 + l16] = x;
    }
}

// ---------------------------------------------------------------------------
extern "C" void kernel_launch(void* const* d_in, const int* in_sizes, int n_in,
                              void* d_out, int out_size, void* d_ws, size_t ws_size,
                              hipStream_t stream) {
  const float* h    = (const float*)d_in[0];   // (8,2048,128) f32
  const int*   adj  = (const int*)  d_in[1];   // (8,2048,2048) i32
  const float* W    = (const float*)d_in[2];   // (128,64) f32
  const float* avec = (const float*)d_in[3];   // (128,1) f32
  float* out = (float*)d_out;                  // (8,2048,64) f32

  // workspace layout: Wh (8*2048*64) | e1 (8*2048) | e2 (8*2048)  ~4.3 MB
  float* Wh = (float*)d_ws;
  float* e1 = Wh + (size_t)GAT_B * GAT_N * GAT_FOUT;
  float* e2 = e1 + (size_t)GAT_B * GAT_N;

  const int waves = GAT_B * GAT_N / 16;        // 1024 i-tiles
  const int wavesPerBlock = 4;                 // 128 threads

  gat_wh_kernel<<<waves / wavesPerBlock, 32 * wavesPerBlock, 0, stream>>>(h, W, Wh);
  gat_e_kernel<<<(GAT_B * GAT_N) / 256, 256, 0, stream>>>(Wh, avec, e1, e2);
  gat_attn_kernel<<<waves / wavesPerBlock, 32 * wavesPerBlock, 0, stream>>>(
      adj, Wh, e1, e2, out);
}